// EstimateAdj_48447231099314
// MI455X (gfx1250) — compile-verified
//
#include <hip/hip_runtime.h>
#include <stdint.h>

#define N   8192
#define ROWS 2                    // rows per workgroup
#define TPB 256                   // 8 waves of 32
#define F4_PER_ROW (N / 4)        // 2048 float4 per row
#define F4_PER_THR (F4_PER_ROW / TPB) // 8 float4 per thread per row

typedef __attribute__((ext_vector_type(2))) float v2f;
typedef __attribute__((ext_vector_type(8))) float v8f;

// ---- CDNA5 async global->LDS copy (gfx1250), inline asm per CDNA5 ISA ----
__device__ __forceinline__ void async_load_b128(uint32_t lds_byte_off, const void* gaddr) {
    asm volatile("global_load_async_to_lds_b128 %0, %1, off"
                 :: "v"(lds_byte_off), "v"((uint64_t)(uintptr_t)gaddr)
                 : "memory");
}
__device__ __forceinline__ void wait_async0() {
    asm volatile("s_wait_asynccnt 0x0" ::: "memory");
}

__global__ __launch_bounds__(TPB)
void normalize_adj_kernel(const float* __restrict__ A,
                          const float* __restrict__ B,
                          float* __restrict__ out) {
    __shared__ float sA[ROWS * N];       // 64 KB
    __shared__ float sB[ROWS * N];       // 64 KB
    __shared__ float part[ROWS * TPB];   // 2 KB partial row sums
    __shared__ float rs[ROWS];           // per-row 1/rowsum

    const int t  = threadIdx.x;
    const int r0 = blockIdx.x * ROWS;

    // ---------- Phase 0: async-stage both row slabs into LDS ----------
    const uint32_t sA_base = (uint32_t)(uintptr_t)(&sA[0]);
    const uint32_t sB_base = (uint32_t)(uintptr_t)(&sB[0]);
    for (int r = 0; r < ROWS; ++r) {
        const size_t grow = (size_t)(r0 + r) * (size_t)N;
        for (int i = 0; i < F4_PER_THR; ++i) {
            const int f4 = i * TPB + t;                       // coalesced float4 index
            const uint32_t off = (uint32_t)((r * N + f4 * 4) * sizeof(float));
            async_load_b128(sA_base + off, A + grow + (size_t)f4 * 4);
            async_load_b128(sB_base + off, B + grow + (size_t)f4 * 4);
        }
    }
    wait_async0();
    __syncthreads();

    // ---------- Phase 1: per-thread partial row sums of A*B ----------
    const float4* sA4 = (const float4*)sA;
    const float4* sB4 = (const float4*)sB;
    for (int r = 0; r < ROWS; ++r) {
        float s = 0.0f;
        for (int i = 0; i < F4_PER_THR; ++i) {
            const int f4 = i * TPB + t;
            float4 a = sA4[r * F4_PER_ROW + f4];
            float4 b = sB4[r * F4_PER_ROW + f4];
            s += a.x * b.x + a.y * b.y + a.z * b.z + a.w * b.w;
        }
        part[r * TPB + t] = s;
    }
    __syncthreads();

    // ---------- Phase 2: wave 0 reduces 2x256 partials with WMMA x ones ----------
    // A-fragment layout (16x4 f32): lane L holds M = L%16, K = 2*(L/16)+{0,1}.
    // Map M -> (row r = M/8, chunk c = M%8); iteration i covers partial
    // indices c*32 + i*4 + K. After 8 accumulating WMMAs, D[M][*] = sum of
    // 32 partials; summing the 8 acc VGPRs gives: lanes 0-15 = row0 total,
    // lanes 16-31 = row1 total (C/D layout: VGPR j = M=j / M=8+j).
    if (t < 32) {
        const int lane  = t;
        const int m     = lane & 15;
        const int rrow  = m >> 3;        // 0..1
        const int chunk = m & 7;         // 0..7
        const int khalf = lane >> 4;     // 0 or 1
        const float* pbase = &part[rrow * TPB + chunk * 32 + 2 * khalf];

        v8f acc = {};
        v2f ones; ones[0] = 1.0f; ones[1] = 1.0f;
        for (int i = 0; i < 8; ++i) {
            v2f a;
            a[0] = pbase[i * 4 + 0];
            a[1] = pbase[i * 4 + 1];
            acc = __builtin_amdgcn_wmma_f32_16x16x4_f32(
                false, a, false, ones, (short)0, acc, false, false);
        }
        float v = acc[0] + acc[1] + acc[2] + acc[3] +
                  acc[4] + acc[5] + acc[6] + acc[7];
        if ((lane & 15) == 0) {
            const float rsum = v + 1.0f;       // + identity diagonal
            float ri = 1.0f / rsum;
            if (__builtin_isinf(ri)) ri = 0.0f;
            rs[lane >> 4] = ri;                // lane 0 -> rs[0], lane 16 -> rs[1]
        }
    }
    __syncthreads();

    // ---------- Phase 3: scale and write out (single global write) ----------
    for (int r = 0; r < ROWS; ++r) {
        const int gr = r0 + r;
        const float ri = rs[r];
        float4* orow = (float4*)(out + (size_t)gr * (size_t)N);
        for (int i = 0; i < F4_PER_THR; ++i) {
            const int f4 = i * TPB + t;
            float4 a = sA4[r * F4_PER_ROW + f4];
            float4 b = sB4[r * F4_PER_ROW + f4];
            const int c0 = f4 * 4;
            float4 o;
            o.x = a.x * b.x + ((c0 + 0 == gr) ? 1.0f : 0.0f);
            o.y = a.y * b.y + ((c0 + 1 == gr) ? 1.0f : 0.0f);
            o.z = a.z * b.z + ((c0 + 2 == gr) ? 1.0f : 0.0f);
            o.w = a.w * b.w + ((c0 + 3 == gr) ? 1.0f : 0.0f);
            o.x *= ri; o.y *= ri; o.z *= ri; o.w *= ri;
            orow[f4] = o;
        }
    }
}

extern "C" void kernel_launch(void* const* d_in, const int* in_sizes, int n_in,
                              void* d_out, int out_size, void* d_ws, size_t ws_size,
                              hipStream_t stream) {
    (void)in_sizes; (void)n_in; (void)out_size; (void)d_ws; (void)ws_size;
    const float* A = (const float*)d_in[0];   // estimated_adj
    const float* B = (const float*)d_in[1];   // ori
    float* out = (float*)d_out;
    dim3 grid(N / ROWS), block(TPB);
    hipLaunchKernelGGL(normalize_adj_kernel, grid, block, 0, stream, A, B, out);
}